// RandomMasking_67113158967612
// MI455X (gfx1250) — compile-verified
//
#include <hip/hip_runtime.h>
#include <hip/hip_bf16.h>

typedef float v4f __attribute__((ext_vector_type(4)));

#define NP_TOT   4096        // 64*64 patches per image
#define NUM_MASK 3072
#define NB       64
#define FLOAT4_PER_ROW 384u  // 512*3/4
#define FLOAT4_PER_IMG 196608u // 512*512*3/4
#define TPB   256
#define NITER 8
#define BLK_F4 (TPB * NITER)   // 2048 float4 per block
#define BLKS_PER_IMG (FLOAT4_PER_IMG / BLK_F4) // 96, exact

// ---------------- mask build ----------------
__global__ void mask_init_kernel(float* __restrict__ maskbuf, int n) {
    int i = blockIdx.x * blockDim.x + threadIdx.x;
    if (i < n) maskbuf[i] = 1.0f;
}

__global__ void mask_scatter_kernel(const int* __restrict__ mi,
                                    float* __restrict__ maskbuf, int total) {
    int i = blockIdx.x * blockDim.x + threadIdx.x;
    if (i < total) {
        int b = i / NUM_MASK;                    // image index
        maskbuf[b * NP_TOT + mi[i]] = 0.0f;      // indices unique per image
    }
}

// ---------------- streaming multiply: 4-deep async global->LDS pipeline ----------------
__device__ __forceinline__ void async_ld_b128(unsigned lds_off, const v4f* gptr) {
    // CDNA5 async copy: per-lane 16B global -> LDS, tracked by ASYNCcnt.
    asm volatile("global_load_async_to_lds_b128 %0, %1, off th:TH_LOAD_NT"
                 :: "v"(lds_off), "v"(gptr) : "memory");
}

__global__ void mask_mul_kernel(const v4f* __restrict__ img,
                                const float* __restrict__ maskbuf,
                                v4f* __restrict__ out) {
    __shared__ v4f tile[4][TPB];
    const unsigned tid = threadIdx.x;
    const unsigned blk = blockIdx.x;

    // Each block lies entirely within one image (96 blocks/image): uniform scalars.
    const unsigned b        = blk / BLKS_PER_IMG;                 // SALU
    const unsigned rem_blk  = (blk - b * BLKS_PER_IMG) * BLK_F4;  // image-relative base
    const float*   maskb    = maskbuf + (b << 12);                // this image's 4096 masks
    const unsigned base     = blk * BLK_F4 + tid;                 // global float4 index

    // low 32 bits of generic address of __shared__ == LDS byte offset
    unsigned lds[4];
#pragma unroll
    for (int j = 0; j < 4; ++j) lds[j] = (unsigned)(size_t)(&tile[j][tid]);

    // prologue: tiles 0..2 in flight
    async_ld_b128(lds[0], img + base);
    async_ld_b128(lds[1], img + base + TPB);
    async_ld_b128(lds[2], img + base + 2 * TPB);

#pragma unroll
    for (int k = 0; k < NITER; ++k) {
        if (k + 3 < NITER) {
            async_ld_b128(lds[(k + 3) & 3], img + base + (unsigned)(k + 3) * TPB);
            asm volatile("s_wait_asynccnt 3" ::: "memory");   // tile k landed
        } else if (k + 2 < NITER) {
            asm volatile("s_wait_asynccnt 2" ::: "memory");
        } else if (k + 1 < NITER) {
            asm volatile("s_wait_asynccnt 1" ::: "memory");
        } else {
            asm volatile("s_wait_asynccnt 0" ::: "memory");
        }

        v4f v = tile[k & 3][tid];   // ds_load_b128, own slot, wave-local ordering only

        // one mask value per aligned float4 (patch col = 24 floats, multiple of 4)
        const unsigned rem = rem_blk + (unsigned)k * TPB + tid;   // < 196608
        const unsigned row = rem / FLOAT4_PER_ROW;                // 0..511
        const unsigned c4  = rem - row * FLOAT4_PER_ROW;          // 0..383
        const float m = maskb[((row >> 3) << 6) + (c4 / 6u)];     // L2-resident

        v *= m;
        __builtin_nontemporal_store(v, &out[base + (unsigned)k * TPB]); // write-once
    }
}

extern "C" void kernel_launch(void* const* d_in, const int* in_sizes, int n_in,
                              void* d_out, int out_size, void* d_ws, size_t ws_size,
                              hipStream_t stream) {
    const float* img = (const float*)d_in[0];       // [64,512,512,3] fp32
    const int*   mi  = (const int*)d_in[1];         // [64,3072] int32
    float* maskbuf   = (float*)d_ws;                // [64,4096] fp32 (1 MB)

    const int mask_n = NB * NP_TOT;                 // 262144
    const int scat_n = NB * NUM_MASK;               // 196608
    mask_init_kernel<<<(mask_n + TPB - 1) / TPB, TPB, 0, stream>>>(maskbuf, mask_n);
    mask_scatter_kernel<<<(scat_n + TPB - 1) / TPB, TPB, 0, stream>>>(mi, maskbuf, scat_n);

    // 12,582,912 float4 = 6144 blocks * 256 threads * 8 iters (exact division)
    const unsigned grid = (NB * FLOAT4_PER_IMG) / BLK_F4;   // 6144
    mask_mul_kernel<<<grid, TPB, 0, stream>>>((const v4f*)img, maskbuf, (v4f*)d_out);
}